// DistanceTransform_25185688224534
// MI455X (gfx1250) — compile-verified
//
#include <hip/hip_runtime.h>

#define EDT_H 512
#define EDT_W 512
#define EDT_N (EDT_H * EDT_W)
#define EDT_INF 524289.0f   // H*H + W*W + 1, exactly representable in fp32

typedef float v2f __attribute__((ext_vector_type(2)));
typedef float v8f __attribute__((ext_vector_type(8)));

// ---------------------------------------------------------------------------
// Scalar scratch init (ws is poisoned 0xAA; re-init every call, graph-safe).
// scal[0] = bits of max(masks), scal[1] = bits of max(d2)
// ---------------------------------------------------------------------------
__global__ void edt_init_scalars(unsigned* scal) {
    if (threadIdx.x == 0) { scal[0] = 0u; scal[1] = 0u; }
}

// ---------------------------------------------------------------------------
// Global max of a non-negative float array via uint-bit atomicMax (exact,
// order-independent -> deterministic).
// ---------------------------------------------------------------------------
__global__ void edt_reduce_max(const float* __restrict__ in,
                               unsigned* __restrict__ outBits) {
    __shared__ float sm[256];
    const int tid = threadIdx.x;
    float m = 0.0f;  // inputs are >= 0
    for (int idx = blockIdx.x * 256 + tid; idx < EDT_N; idx += gridDim.x * 256)
        m = fmaxf(m, in[idx]);
    sm[tid] = m;
    __syncthreads();
    for (int s = 128; s > 0; s >>= 1) {
        if (tid < s) sm[tid] = fmaxf(sm[tid], sm[tid + s]);
        __syncthreads();
    }
    if (tid == 0) atomicMax(outBits, __float_as_uint(sm[0]));
}

// ---------------------------------------------------------------------------
// 512x512 f32 transpose through the matrix pipe. One wave32 per 16x16 tile.
//   D = sum_k A_k * B_k,  A_k(m,kk) = T(4k+kk, m)  (row-segment loads of T),
//   B_k = I[4k:4k+4, :]   (identity slice built in registers)
// => D = T^T in C-layout, stored to the transposed tile location.
// A layout (16x4 f32): lane L holds M=L%16, K = 2*(L/16) + vgpr.
// B layout (4x16 f32): lane L holds N=L%16, row K = 2*(L/16) + vgpr.
// C layout (16x16 f32): lane L holds N=L%16, M = vgpr + 8*(L/16).
// binarize!=0 applies f = (x==0 ? 0 : INF) during the load (pass-1 input map).
// ---------------------------------------------------------------------------
__global__ void edt_wmma_transpose(const float* __restrict__ src,
                                   float* __restrict__ dst, int binarize) {
    const int lane = threadIdx.x & 31;
    const int tile = blockIdx.x * 8 + (threadIdx.x >> 5);  // 1024 tiles
    const int tr = tile >> 5;        // tile row  (32 tiles / dim)
    const int tc = tile & 31;        // tile col
    const int c    = lane & 15;
    const int half = lane >> 4;

    v8f acc = {};
#pragma unroll
    for (int k = 0; k < 4; ++k) {
        const int r0 = tr * 16 + 4 * k + 2 * half;
        float x0 = src[(size_t)r0 * EDT_W + tc * 16 + c];
        float x1 = src[(size_t)(r0 + 1) * EDT_W + tc * 16 + c];
        if (binarize) {
            x0 = (x0 == 0.0f) ? 0.0f : EDT_INF;
            x1 = (x1 == 0.0f) ? 0.0f : EDT_INF;
        }
        v2f a, b;
        a.x = x0;
        a.y = x1;
        b.x = (c == 4 * k + 2 * half) ? 1.0f : 0.0f;        // identity row 4k+2*half
        b.y = (c == 4 * k + 2 * half + 1) ? 1.0f : 0.0f;    // identity row 4k+2*half+1
        acc = __builtin_amdgcn_wmma_f32_16x16x4_f32(
            /*neg_a=*/false, a, /*neg_b=*/false, b,
            /*c_mod=*/(short)0, acc, /*reuse_a=*/false, /*reuse_b=*/false);
    }
#pragma unroll
    for (int v = 0; v < 8; ++v)
        dst[(size_t)(tc * 16 + v + 8 * half) * EDT_W + tr * 16 + c] = acc[v];
}

// ---------------------------------------------------------------------------
// Row-wise min-plus: out[r][i] = min_j (i-j)^2 + in[r][j].
// One block per row; row staged to LDS; in-place safe (full row read before
// any write). All arithmetic is exact fp32 integer math (< 2^24).
// ---------------------------------------------------------------------------
__global__ void edt_minplus(const float* __restrict__ in,
                            float* __restrict__ out) {
    __shared__ float row[EDT_W];
    const int r = blockIdx.x;
    const int t = threadIdx.x;
    row[t]       = in[(size_t)r * EDT_W + t];
    row[t + 256] = in[(size_t)r * EDT_W + t + 256];
    __syncthreads();

    const float i0 = (float)t, i1 = (float)(t + 256);
    float m0 = 3.0e38f, m1 = 3.0e38f;
#pragma unroll 8
    for (int j = 0; j < EDT_W; ++j) {
        const float fj = row[j];
        const float d0 = i0 - (float)j;
        const float d1 = i1 - (float)j;
        m0 = fminf(m0, fmaf(d0, d0, fj));
        m1 = fminf(m1, fmaf(d1, d1, fj));
    }
    out[(size_t)r * EDT_W + t]       = m0;
    out[(size_t)r * EDT_W + t + 256] = m1;
}

// Same pass, but also emits sqrt(d2) and reduces max(d2) (sqrt is monotone,
// so sqrt(max d2) == max(sqrt d2)).
__global__ void edt_minplus_sqrt_max(const float* __restrict__ in,
                                     float* __restrict__ out,
                                     unsigned* __restrict__ maxBits) {
    __shared__ float row[EDT_W];
    __shared__ float sm[256];
    const int r = blockIdx.x;
    const int t = threadIdx.x;
    row[t]       = in[(size_t)r * EDT_W + t];
    row[t + 256] = in[(size_t)r * EDT_W + t + 256];
    __syncthreads();

    const float i0 = (float)t, i1 = (float)(t + 256);
    float m0 = 3.0e38f, m1 = 3.0e38f;
#pragma unroll 8
    for (int j = 0; j < EDT_W; ++j) {
        const float fj = row[j];
        const float d0 = i0 - (float)j;
        const float d1 = i1 - (float)j;
        m0 = fminf(m0, fmaf(d0, d0, fj));
        m1 = fminf(m1, fmaf(d1, d1, fj));
    }
    out[(size_t)r * EDT_W + t]       = sqrtf(m0);
    out[(size_t)r * EDT_W + t + 256] = sqrtf(m1);

    sm[t] = fmaxf(m0, m1);
    __syncthreads();
    for (int s = 128; s > 0; s >>= 1) {
        if (t < s) sm[t] = fmaxf(sm[t], sm[t + s]);
        __syncthreads();
    }
    if (t == 0) atomicMax(maxBits, __float_as_uint(sm[0]));
}

// ---------------------------------------------------------------------------
// skeleton = dis / max(dis);  boundary = masks/max(masks) - skeleton.
// dis lives in outBd's slot (read before overwrite, same idx -> safe).
// ---------------------------------------------------------------------------
__global__ void edt_finalize(const float* __restrict__ masks,
                             const float* __restrict__ dis,
                             float* __restrict__ outSk,
                             float* __restrict__ outBd,
                             const unsigned* __restrict__ scal) {
    const int idx = blockIdx.x * 256 + threadIdx.x;
    const float maxM = __uint_as_float(scal[0]);
    const float maxD = sqrtf(__uint_as_float(scal[1]));
    const float d  = dis[idx];
    const float sk = d / maxD;
    const float bd = masks[idx] / maxM - sk;
    outSk[idx] = sk;
    outBd[idx] = bd;
}

extern "C" void kernel_launch(void* const* d_in, const int* in_sizes, int n_in,
                              void* d_out, int out_size, void* d_ws, size_t ws_size,
                              hipStream_t stream) {
    (void)in_sizes; (void)n_in; (void)out_size; (void)ws_size;
    const float* masks = (const float*)d_in[0];
    float* out   = (float*)d_out;
    float* half0 = out;             // scratch, becomes skeleton
    float* half1 = out + EDT_N;     // scratch, becomes boundary
    unsigned* scal = (unsigned*)d_ws;

    edt_init_scalars<<<1, 32, 0, stream>>>(scal);
    edt_reduce_max<<<256, 256, 0, stream>>>(masks, scal + 0);
    // pass 1 (down columns): transpose(f(masks)) -> half0, row min-plus in place
    edt_wmma_transpose<<<128, 256, 0, stream>>>(masks, half0, /*binarize=*/1);
    edt_minplus<<<EDT_H, 256, 0, stream>>>(half0, half0);
    // pass 2 (along rows): transpose back -> half1, row min-plus + sqrt + max
    edt_wmma_transpose<<<128, 256, 0, stream>>>(half0, half1, /*binarize=*/0);
    edt_minplus_sqrt_max<<<EDT_H, 256, 0, stream>>>(half1, half1, scal + 1);
    // normalize + boundary
    edt_finalize<<<EDT_N / 256, 256, 0, stream>>>(masks, half1, half0, half1, scal);
}